// SELayer_8203387535782
// MI455X (gfx1250) — compile-verified
//
#include <hip/hip_runtime.h>
#include <hip/hip_bf16.h>

typedef float v2f __attribute__((ext_vector_type(2)));
typedef float v4f __attribute__((ext_vector_type(4)));
typedef float v8f __attribute__((ext_vector_type(8)));

#define SE_EPS 1e-5f
#define NB   32     // batch
#define NC   256    // channels
#define NCS  64     // squeezed channels
#define NHW  3136   // 56*56
#define NHW4 784    // NHW/4

// ---------------------------------------------------------------------------
// Kernel 1: squeeze = global average pool. One block per (b,c) plane.
// Plain RT loads so x becomes resident in the 192MB L2 for the excite pass.
// ---------------------------------------------------------------------------
__global__ __launch_bounds__(256) void se_squeeze(const float* __restrict__ x,
                                                  float* __restrict__ s) {
    __shared__ float partial[8];
    const int plane = blockIdx.x;                       // = b*NC + c
    const v4f* xp = (const v4f*)(x + (size_t)plane * NHW);
    float sum = 0.f;
    for (int i = threadIdx.x; i < NHW4; i += 256) {
        v4f v = xp[i];
        sum += (v.x + v.y) + (v.z + v.w);
    }
    // wave32 reduction
    for (int off = 16; off >= 1; off >>= 1)
        sum += __shfl_down(sum, off, 32);
    const int wave = threadIdx.x >> 5;
    if ((threadIdx.x & 31) == 0) partial[wave] = sum;
    __syncthreads();
    if (threadIdx.x == 0) {
        float t = 0.f;
        for (int w = 0; w < 8; ++w) t += partial[w];
        s[plane] = t * (1.0f / (float)NHW);
    }
}

// ---------------------------------------------------------------------------
// Kernel 2: gate = hsig(BN2(relu(BN1(s*w1^T)) * w2^T)), fp32 WMMA 16x16x4.
// One workgroup, 8 waves. GEMM1: one 16x16 tile/wave (M=32,N=64,K=256).
// GEMM2: 4 tiles/wave (M=32,N=256,K=64).
// ---------------------------------------------------------------------------
__global__ __launch_bounds__(256) void se_gate(
    const float* __restrict__ s,        // [NB][NC] means
    const float* __restrict__ w1,       // [NCS][NC]
    const float* __restrict__ g1, const float* __restrict__ b1,
    const float* __restrict__ m1, const float* __restrict__ vr1,
    const float* __restrict__ w2,       // [NC][NCS]
    const float* __restrict__ g2, const float* __restrict__ b2,
    const float* __restrict__ m2, const float* __restrict__ vr2,
    float* __restrict__ gate)           // [NB][NC]
{
    __shared__ float hlds[NB][NCS];
    const int wave = threadIdx.x >> 5;
    const int lane = threadIdx.x & 31;
    const int half = lane >> 4;         // lane half selects K pair (ISA 7.12.2)
    const int lid  = lane & 15;

    // ---- GEMM1: h[32x64] = s[32x256] x w1t[256x64] ----
    {
        const int mt = wave >> 2;            // 0..1  (M tile)
        const int nt = wave & 3;             // 0..3  (N tile)
        const int mA = mt * 16 + lid;        // A-frag row (batch)
        const int nB = nt * 16 + lid;        // B-frag col (channel of h)
        v8f acc = {};
        for (int k0 = 0; k0 < NC; k0 += 4) {
            const int k = k0 + 2 * half;
            v2f a = *(const v2f*)(s  + mA * NC + k);   // A[m][k], A[m][k+1]
            v2f b = *(const v2f*)(w1 + nB * NC + k);   // Bt[k][n], Bt[k+1][n]
            acc = __builtin_amdgcn_wmma_f32_16x16x4_f32(
                false, a, false, b, (short)0, acc, false, false);
        }
        // BN1 + ReLU (per-channel n = nB, lane-constant)
        const float sc = g1[nB] * rsqrtf(vr1[nB] + SE_EPS);
        const float bi = b1[nB] - m1[nB] * sc;
        #pragma unroll
        for (int r = 0; r < 8; ++r) {
            const int m = mt * 16 + r + 8 * half;      // D layout: M=r / r+8
            float h = acc[r] * sc + bi;
            hlds[m][nB] = h > 0.f ? h : 0.f;
        }
    }
    __syncthreads();

    // ---- GEMM2: g[32x256] = h[32x64] x w2t[64x256] ----
    {
        const int mt = wave >> 2;
        const int mA = mt * 16 + lid;
        #pragma unroll
        for (int t = 0; t < 4; ++t) {
            const int nt = (wave & 3) * 4 + t;         // 0..15
            const int nB = nt * 16 + lid;              // output channel c
            v8f acc = {};
            for (int k0 = 0; k0 < NCS; k0 += 4) {
                const int k = k0 + 2 * half;
                v2f a = *(const v2f*)(&hlds[mA][k]);
                v2f b = *(const v2f*)(w2 + nB * NCS + k);  // w2t[k][c]=w2[c][k]
                acc = __builtin_amdgcn_wmma_f32_16x16x4_f32(
                    false, a, false, b, (short)0, acc, false, false);
            }
            const float sc = g2[nB] * rsqrtf(vr2[nB] + SE_EPS);
            const float bi = b2[nB] - m2[nB] * sc;
            #pragma unroll
            for (int r = 0; r < 8; ++r) {
                const int m = mt * 16 + r + 8 * half;
                float g = acc[r] * sc + bi;
                g = fminf(fmaxf(g + 3.f, 0.f), 6.f) * (1.f / 6.f);
                gate[m * NC + nB] = g;
            }
        }
    }
}

// ---------------------------------------------------------------------------
// Kernel 3: excite = x * gate[b][c]. One v4f per thread; x read hits L2
// (resident from kernel 1); output uses non-temporal stores so the 103MB of
// stores do not evict x from L2.
// ---------------------------------------------------------------------------
__global__ __launch_bounds__(256) void se_excite(const float* __restrict__ x,
                                                 const float* __restrict__ gate,
                                                 float* __restrict__ out) {
    const unsigned i = blockIdx.x * 256u + threadIdx.x;  // float4 index
    const unsigned plane = i / NHW4;                     // = b*NC + c
    const float g = gate[plane];
    v4f v = ((const v4f*)x)[i];
    v *= g;
    __builtin_nontemporal_store(v, ((v4f*)out) + i);
}

extern "C" void kernel_launch(void* const* d_in, const int* in_sizes, int n_in,
                              void* d_out, int out_size, void* d_ws, size_t ws_size,
                              hipStream_t stream) {
    (void)in_sizes; (void)n_in; (void)out_size; (void)ws_size;

    const float* x   = (const float*)d_in[0];
    const float* w1  = (const float*)d_in[1];
    const float* g1  = (const float*)d_in[2];
    const float* b1  = (const float*)d_in[3];
    const float* m1  = (const float*)d_in[4];
    const float* vr1 = (const float*)d_in[5];
    const float* w2  = (const float*)d_in[6];
    const float* g2  = (const float*)d_in[7];
    const float* b2  = (const float*)d_in[8];
    const float* m2  = (const float*)d_in[9];
    const float* vr2 = (const float*)d_in[10];
    float* out = (float*)d_out;

    float* s_ws    = (float*)d_ws;            // NB*NC floats (means)
    float* gate_ws = s_ws + NB * NC;          // NB*NC floats (gates)

    se_squeeze<<<NB * NC, 256, 0, stream>>>(x, s_ws);
    se_gate<<<1, 256, 0, stream>>>(s_ws, w1, g1, b1, m1, vr1,
                                   w2, g2, b2, m2, vr2, gate_ws);
    const int n4 = NB * NC * NHW4;            // 6,422,528 float4s
    se_excite<<<n4 / 256, 256, 0, stream>>>(x, gate_ws, out);
}